// PFAE_68539088110349
// MI455X (gfx1250) — compile-verified
//
#include <hip/hip_runtime.h>
#include <hip/hip_bf16.h>

typedef __attribute__((ext_vector_type(2))) float v2f;
typedef __attribute__((ext_vector_type(8))) float v8f;

#define THREADS 256

// ---------------- degree / normalization ----------------

// deg initialized to 1.0 (self-loop contribution)
__global__ void k_deg_init(float* __restrict__ deg, int N) {
  int n = blockIdx.x * blockDim.x + threadIdx.x;
  if (n < N) deg[n] = 1.0f;
}

// deg[dst] += 1 for every real edge (edge_index is int64; dst = second half)
__global__ void k_deg_edges(const long long* __restrict__ ei, float* __restrict__ deg, int E) {
  int e = blockIdx.x * blockDim.x + threadIdx.x;
  if (e < E) {
    int d = (int)ei[(size_t)E + e];
    atomicAdd(&deg[d], 1.0f);
  }
}

// dinv = rsqrt(deg), in place
__global__ void k_rsqrt(float* __restrict__ dd, int N) {
  int n = blockIdx.x * blockDim.x + threadIdx.x;
  if (n < N) dd[n] = rsqrtf(dd[n]);
}

// ---------------- dense transforms via V_WMMA_F32_16X16X4_F32 ----------------
// One wave32 computes a 16(node)x16(col) f32 tile; K accumulated in steps of 4.
// A (16x4 f32) layout: lanes 0-15 row M=lane hold K=k0,k0+1; lanes 16-31 hold K=k0+2,k0+3.
// B (4x16 f32) layout: mirrored (row K striped across lanes 0-15 / 16-31).
// C/D: VGPR i = row (i + 8*hi), col = lane&15.
// B zero-padding uses a clamped (in-bounds) address + v_cndmask so the weight
// loads stay unconditional (no EXEC-masked branch regions between WMMAs).

// xw1 = x @ W1   (x: N x 16, W1: 16 x 8 zero-padded to 16 cols)
__global__ void __launch_bounds__(THREADS)
k_xw1_wmma(const float* __restrict__ x, const float* __restrict__ W1,
           float* __restrict__ xw1, int N) {
  const int wave = (int)((blockIdx.x * (unsigned)blockDim.x + threadIdx.x) >> 5);
  const int lane = threadIdx.x & 31;
  const int base = wave * 16;
  if (base >= N) return;                 // whole-wave uniform: EXEC stays all-1s for WMMA
  const int  m   = lane & 15;
  const int  hi  = lane >> 4;
  const bool inc = (m < 8);              // valid output column?
  const int  mc  = inc ? m : 0;          // clamped, always-in-bounds column
  const float* xrow = x + (size_t)(base + m) * 16;
  v8f acc = {};
#pragma unroll
  for (int k0 = 0; k0 < 16; k0 += 4) {
    const int ka = k0 + hi * 2;
    v2f a; a.x = xrow[ka]; a.y = xrow[ka + 1];
    float w0 = W1[ka * 8 + mc];          // unconditional loads
    float w1 = W1[(ka + 1) * 8 + mc];
    v2f b;
    b.x = inc ? w0 : 0.0f;               // lowers to v_cndmask, no branch
    b.y = inc ? w1 : 0.0f;
    acc = __builtin_amdgcn_wmma_f32_16x16x4_f32(false, a, false, b, (short)0, acc,
                                                false, false);
  }
  if (inc) {
#pragma unroll
    for (int i = 0; i < 8; ++i)
      xw1[(size_t)(base + i + hi * 8) * 8 + m] = acc[i];
  }
}

// xw2 = relu(agg1 + b1) @ W2   (fused bias+ReLU in the A load; W2: 8 x 4 padded)
__global__ void __launch_bounds__(THREADS)
k_xw2_wmma(const float* __restrict__ agg1, const float* __restrict__ b1,
           const float* __restrict__ W2, float* __restrict__ xw2, int N) {
  const int wave = (int)((blockIdx.x * (unsigned)blockDim.x + threadIdx.x) >> 5);
  const int lane = threadIdx.x & 31;
  const int base = wave * 16;
  if (base >= N) return;
  const int  m   = lane & 15;
  const int  hi  = lane >> 4;
  const bool inc = (m < 4);
  const int  mc  = inc ? m : 0;
  const float* hrow = agg1 + (size_t)(base + m) * 8;
  v8f acc = {};
#pragma unroll
  for (int k0 = 0; k0 < 8; k0 += 4) {
    const int ka = k0 + hi * 2;
    v2f a;
    a.x = fmaxf(hrow[ka]     + b1[ka],     0.0f);
    a.y = fmaxf(hrow[ka + 1] + b1[ka + 1], 0.0f);
    float w0 = W2[ka * 4 + mc];
    float w1 = W2[(ka + 1) * 4 + mc];
    v2f b;
    b.x = inc ? w0 : 0.0f;
    b.y = inc ? w1 : 0.0f;
    acc = __builtin_amdgcn_wmma_f32_16x16x4_f32(false, a, false, b, (short)0, acc,
                                                false, false);
  }
  if (inc) {
#pragma unroll
    for (int i = 0; i < 8; ++i)
      xw2[(size_t)(base + i + hi * 8) * 4 + m] = acc[i];
  }
}

// ---------------- self-loop seed + edge scatter ----------------

// agg1 = xw1 * dinv^2   (self-loop term; also initializes poisoned scratch)
__global__ void k_seed1(const float* __restrict__ xw1, const float* __restrict__ dinv,
                        float* __restrict__ agg1, int N) {
  int i = blockIdx.x * blockDim.x + threadIdx.x;
  if (i < N * 8) {
    float di = dinv[i >> 3];
    agg1[i] = xw1[i] * di * di;
  }
}

// out = xw2 * dinv^2 + b2   (self-loop term + final bias; initializes d_out)
__global__ void k_seed2(const float* __restrict__ xw2, const float* __restrict__ dinv,
                        const float* __restrict__ b2, float* __restrict__ out, int N) {
  int i = blockIdx.x * blockDim.x + threadIdx.x;
  if (i < N * 4) {
    float di = dinv[i >> 2];
    out[i] = xw2[i] * di * di + b2[i & 3];
  }
}

// layer-1 edge scatter: agg1[dst] += xw1[src] * dinv[src]*dinv[dst]  (8 ch)
__global__ void k_scatter1(const long long* __restrict__ ei, const float* __restrict__ dinv,
                           const float* __restrict__ xw1, float* __restrict__ agg1, int E) {
  int e = blockIdx.x * blockDim.x + threadIdx.x;
  if (e >= E) return;
  int s = (int)ei[e];
  int d = (int)ei[(size_t)E + e];
  float nrm = dinv[s] * dinv[d];
  const float4* xs = (const float4*)(xw1 + (size_t)s * 8);
  float4 v0 = xs[0], v1 = xs[1];
  float* o = agg1 + (size_t)d * 8;
  atomicAdd(o + 0, v0.x * nrm); atomicAdd(o + 1, v0.y * nrm);
  atomicAdd(o + 2, v0.z * nrm); atomicAdd(o + 3, v0.w * nrm);
  atomicAdd(o + 4, v1.x * nrm); atomicAdd(o + 5, v1.y * nrm);
  atomicAdd(o + 6, v1.z * nrm); atomicAdd(o + 7, v1.w * nrm);
}

// layer-2 edge scatter: out[dst] += xw2[src] * dinv[src]*dinv[dst]  (4 ch)
__global__ void k_scatter2(const long long* __restrict__ ei, const float* __restrict__ dinv,
                           const float* __restrict__ xw2, float* __restrict__ out, int E) {
  int e = blockIdx.x * blockDim.x + threadIdx.x;
  if (e >= E) return;
  int s = (int)ei[e];
  int d = (int)ei[(size_t)E + e];
  float nrm = dinv[s] * dinv[d];
  float4 v = *(const float4*)(xw2 + (size_t)s * 4);
  float* o = out + (size_t)d * 4;
  atomicAdd(o + 0, v.x * nrm); atomicAdd(o + 1, v.y * nrm);
  atomicAdd(o + 2, v.z * nrm); atomicAdd(o + 3, v.w * nrm);
}

// ---------------- launcher ----------------

extern "C" void kernel_launch(void* const* d_in, const int* in_sizes, int n_in,
                              void* d_out, int out_size, void* d_ws, size_t ws_size,
                              hipStream_t stream) {
  const float*     x  = (const float*)d_in[0];
  const long long* ei = (const long long*)d_in[1];   // int64 per reference
  const float*     W1 = (const float*)d_in[2];
  const float*     b1 = (const float*)d_in[3];
  const float*     W2 = (const float*)d_in[4];
  const float*     b2 = (const float*)d_in[5];
  float*           out = (float*)d_out;

  const int N = in_sizes[0] / 16;   // IN_CH = 16
  const int E = in_sizes[1] / 2;    // edge_index is (2, E)

  // scratch layout (all L2-resident): dinv | xw1 | agg1 | xw2  -> ~42 MB total
  float* dinv = (float*)d_ws;                 // N
  float* xw1  = dinv + (size_t)N;             // N*8
  float* agg1 = xw1  + (size_t)N * 8;         // N*8
  float* xw2  = agg1 + (size_t)N * 8;         // N*4

  const int gN  = (N + THREADS - 1) / THREADS;
  const int gE  = (E + THREADS - 1) / THREADS;
  const int gN8 = (N * 8 + THREADS - 1) / THREADS;
  const int gN4 = (N * 4 + THREADS - 1) / THREADS;
  const int tiles = (N + 15) / 16;                              // one wave32 per 16-node tile
  const int gW  = (tiles + (THREADS / 32) - 1) / (THREADS / 32);

  k_deg_init<<<gN,  THREADS, 0, stream>>>(dinv, N);
  k_deg_edges<<<gE, THREADS, 0, stream>>>(ei, dinv, E);
  k_rsqrt<<<gN,     THREADS, 0, stream>>>(dinv, N);

  k_xw1_wmma<<<gW,  THREADS, 0, stream>>>(x, W1, xw1, N);
  k_seed1<<<gN8,    THREADS, 0, stream>>>(xw1, dinv, agg1, N);
  k_scatter1<<<gE,  THREADS, 0, stream>>>(ei, dinv, xw1, agg1, E);

  k_xw2_wmma<<<gW,  THREADS, 0, stream>>>(agg1, b1, W2, xw2, N);
  k_seed2<<<gN4,    THREADS, 0, stream>>>(xw2, dinv, b2, out, N);
  k_scatter2<<<gE,  THREADS, 0, stream>>>(ei, dinv, xw2, out, E);
}